// ObliviousDecisionTree_55473797595541
// MI455X (gfx1250) — compile-verified
//
#include <hip/hip_runtime.h>
#include <hip/hip_bf16.h>

// NODE (oblivious decision tree ensemble) forward, fused for MI455X (gfx1250).
//   BATCH=2048, INPUT_DIM=256, NUM_TREES=256, DEPTH=6, TREE_DIM=2
//
// Pipeline:
//  K1: per-(tree,depth) sparsemax (bitonic sort + scan in LDS) -> padded
//      selector matrix sel[K=256][NCOL=2048] (8 cols/tree, pads zeroed) +
//      per-column scale = exp(-log_temp), bias = -thresh*scale.
//  K2: fused GEMM (V_WMMA_F32_16X16X4_F32, exact f32) + sparsemoid +
//      leaf-weight product + response dot. Feature-value intermediate
//      ([2048x1536], 12MB) never leaves LDS/VGPRs.

typedef float v2f __attribute__((ext_vector_type(2)));
typedef float v8f __attribute__((ext_vector_type(8)));

#define NUM_TREES 256
#define DEPTH     6
#define TREE_DIM  2
#define INPUT_DIM 256
#define BATCH     2048
#define PAD_D     8                     // pad 6 depths -> 8 so 16-col WMMA tile = 2 trees
#define NCOL      (NUM_TREES * PAD_D)   // 2048 padded columns
#define NREAL     (NUM_TREES * DEPTH)   // 1536 real columns
#define LDSA_STRIDE 260                 // 256 + 4 pad -> conflict-free A-fragment reads

// ---------------------------------------------------------------------------
// Kernel 1: sparsemax per (tree,depth) column + fold thresholds/temperatures.
// One 256-thread block per padded column (2048 blocks; d>=6 blocks write zeros).
// ---------------------------------------------------------------------------
__global__ void __launch_bounds__(INPUT_DIM)
sparsemax_kernel(const float* __restrict__ fsl,     // [INPUT_DIM][NUM_TREES][DEPTH]
                 const float* __restrict__ ft,      // [NUM_TREES][DEPTH]
                 const float* __restrict__ lt,      // [NUM_TREES][DEPTH]
                 float* __restrict__ sel,           // [INPUT_DIM][NCOL]
                 float* __restrict__ scaleb)        // [2][NCOL]
{
    __shared__ float z[INPUT_DIM];    // sorted (descending)
    __shared__ float cz[INPUT_DIM];   // inclusive cumsum of z
    __shared__ float red[INPUT_DIM];  // reduction buffer

    const int tid = threadIdx.x;
    const int jp  = blockIdx.x;       // padded column
    const int n   = jp >> 3;
    const int d   = jp & 7;

    if (d >= DEPTH) {                 // pad column: zero so GEMM ignores it
        sel[tid * NCOL + jp] = 0.0f;
        if (tid == 0) { scaleb[jp] = 0.0f; scaleb[NCOL + jp] = 0.0f; }
        return;
    }

    const float x = fsl[tid * NREAL + n * DEPTH + d];
    z[tid] = x;
    __syncthreads();

    // Bitonic sort, descending.
    for (int k = 2; k <= INPUT_DIM; k <<= 1) {
        for (int j = k >> 1; j > 0; j >>= 1) {
            const int ixj = tid ^ j;
            if (ixj > tid) {
                const float a = z[tid], b = z[ixj];
                const bool desc = ((tid & k) == 0);
                if (desc ? (a < b) : (a > b)) { z[tid] = b; z[ixj] = a; }
            }
            __syncthreads();
        }
    }

    // Inclusive scan (Hillis-Steele).
    cz[tid] = z[tid];
    __syncthreads();
    for (int off = 1; off < INPUT_DIM; off <<= 1) {
        const float add = (tid >= off) ? cz[tid - off] : 0.0f;
        __syncthreads();
        cz[tid] += add;
        __syncthreads();
    }

    // Support size k_sel = sum[(1 + k*z_k) > cz_k]  (support is a prefix).
    red[tid] = ((1.0f + (float)(tid + 1) * z[tid]) > cz[tid]) ? 1.0f : 0.0f;
    __syncthreads();
    for (int s = INPUT_DIM >> 1; s > 0; s >>= 1) {
        if (tid < s) red[tid] += red[tid + s];
        __syncthreads();
    }
    const int   ksel = (int)red[0];
    const float tau  = (cz[ksel - 1] - 1.0f) / (float)ksel;

    sel[tid * NCOL + jp] = fmaxf(x - tau, 0.0f);

    if (tid == 0) {
        const float s = expf(-lt[n * DEPTH + d]);
        scaleb[jp]        = s;                          // tl = fv*s + b
        scaleb[NCOL + jp] = -ft[n * DEPTH + d] * s;
    }
}

// ---------------------------------------------------------------------------
// Kernel 2: fused GEMM (WMMA f32 16x16x4) + sparsemoid + leaf product + dot.
// Block = 128 threads (4 waves). Tile = 16 batch rows x 8 trees (64 pad cols).
// Wave w owns the 16x16 WMMA tile covering padded columns [w*16, w*16+16).
// ---------------------------------------------------------------------------
__global__ void __launch_bounds__(128)
node_forward_kernel(const float* __restrict__ inputs,  // [BATCH][INPUT_DIM]
                    const float* __restrict__ sel,     // [INPUT_DIM][NCOL]
                    const float* __restrict__ scaleb,  // [2][NCOL]
                    const float* __restrict__ resp,    // [NUM_TREES][TREE_DIM][64]
                    float* __restrict__ out)           // [BATCH][NUM_TREES*TREE_DIM]
{
    __shared__ float ldsA[16 * LDSA_STRIDE];  // A tile: 16 rows x 256 K (padded stride)
    __shared__ float ldsC[16 * 64];           // sparsemoid(tl): 16 rows x 64 padded cols

    const int tid   = threadIdx.x;
    const int lane  = tid & 31;
    const int wave  = tid >> 5;               // 0..3
    const int m0    = blockIdx.x * 16;        // batch-row base
    const int tree0 = blockIdx.y * 8;         // tree base
    const int col0  = tree0 * PAD_D;          // padded-column base

    // --- Stage A tile (16 x 256 f32) into LDS, float4 coalesced ---
    {
        const float4* src = (const float4*)(inputs + m0 * INPUT_DIM);
        for (int i = tid; i < 16 * (INPUT_DIM / 4); i += 128) {
            const int row = i >> 6;           // / (256/4)
            const int k4  = i & 63;
            const float4 v = src[row * 64 + k4];
            float* dst = &ldsA[row * LDSA_STRIDE + k4 * 4];
            dst[0] = v.x; dst[1] = v.y; dst[2] = v.z; dst[3] = v.w;
        }
    }
    __syncthreads();

    // --- GEMM: acc(16x16) += A(16x4) * B(4x16), K = 256 in steps of 4 ---
    // A layout: lanes 0-15 & 16-31 both M=0..15; VGPR0={K0|K2}, VGPR1={K1|K3}.
    // B layout: 16 N per half-wave;            VGPR0={K0|K1}, VGPR1={K2|K3}.
    const int half = lane >> 4;               // 0 / 1
    const int nl   = lane & 15;
    const int ncol = col0 + wave * 16 + nl;   // this lane's output column
    const float* aBase = &ldsA[nl * LDSA_STRIDE];
    const float* bBase = sel + ncol;

    v8f acc = {0.f, 0.f, 0.f, 0.f, 0.f, 0.f, 0.f, 0.f};
    #pragma unroll 8
    for (int kk = 0; kk < INPUT_DIM; kk += 4) {
        v2f a, b;
        const int ka = kk + half * 2;
        a.x = aBase[ka];
        a.y = aBase[ka + 1];
        const int kb = kk + half;
        b.x = bBase[(size_t)kb * NCOL];
        b.y = bBase[(size_t)(kb + 2) * NCOL];
        acc = __builtin_amdgcn_wmma_f32_16x16x4_f32(
            /*neg_a=*/false, a, /*neg_b=*/false, b,
            /*c_mod=*/(short)0, acc, /*reuse_a=*/false, /*reuse_b=*/false);
    }

    // --- sparsemoid: c1 = clamp(0.5*(fv*scale+bias)+0.5, 0, 1) -> LDS ---
    // C/D layout: VGPR v, lane l -> (M = v + 8*half, N = nl).
    const float s  = scaleb[ncol];
    const float bb = scaleb[NCOL + ncol];
    #pragma unroll
    for (int v = 0; v < 8; v++) {
        const float tl = fmaf(acc[v], s, bb);
        const float c1 = fminf(fmaxf(fmaf(0.5f, tl, 0.5f), 0.0f), 1.0f);
        ldsC[(v + half * 8) * 64 + wave * 16 + nl] = c1;
    }
    __syncthreads();

    // --- Leaf phase: one thread per (row, local tree); 16*8 = 128 threads ---
    const int row = tid >> 3;                 // 0..15
    const int t   = tid & 7;                  // 0..7
    const float* c1p = &ldsC[row * 64 + t * PAD_D];

    float c1v[DEPTH], c0v[DEPTH];
    #pragma unroll
    for (int d = 0; d < DEPTH; d++) {
        c1v[d] = c1p[d];                      // sparsemoid(+tl): bit==0 factor
        c0v[d] = 1.0f - c1v[d];               // sparsemoid(-tl): bit==1 factor
    }

    // Leaf weights via doubling: 126 muls, fully unrolled -> stays in VGPRs.
    float w[64];
    w[0] = 1.0f;
    #pragma unroll
    for (int d = 0; d < DEPTH; d++) {
        const int hs = 1 << d;
        #pragma unroll
        for (int c = hs - 1; c >= 0; c--) {
            const float base = w[c];
            w[c + hs] = base * c0v[d];        // bit d == 1
            w[c]      = base * c1v[d];        // bit d == 0
        }
    }

    // out[b, tree, :] = w . response[tree, :, :]
    const int tree = tree0 + t;
    const float* r0 = resp + tree * (TREE_DIM * 64);
    const float* r1 = r0 + 64;
    float a0 = 0.0f, a1 = 0.0f;
    #pragma unroll
    for (int c = 0; c < 64; c++) {
        a0 = fmaf(w[c], r0[c], a0);
        a1 = fmaf(w[c], r1[c], a1);
    }
    const int brow = m0 + row;
    out[brow * (NUM_TREES * TREE_DIM) + tree * 2 + 0] = a0;
    out[brow * (NUM_TREES * TREE_DIM) + tree * 2 + 1] = a1;
}

// ---------------------------------------------------------------------------
extern "C" void kernel_launch(void* const* d_in, const int* in_sizes, int n_in,
                              void* d_out, int out_size, void* d_ws, size_t ws_size,
                              hipStream_t stream) {
    const float* inputs = (const float*)d_in[0];   // [2048][256]
    const float* fsl    = (const float*)d_in[1];   // [256][256][6]
    const float* ft     = (const float*)d_in[2];   // [256][6]
    const float* lt     = (const float*)d_in[3];   // [256][6]
    const float* resp   = (const float*)d_in[4];   // [256][2][64]
    float* out = (float*)d_out;                    // [2048][512]

    // Workspace: sel (256*2048 f32 = 2MB) + scale/bias (2*2048 f32).
    float* sel    = (float*)d_ws;
    float* scaleb = sel + (size_t)INPUT_DIM * NCOL;

    sparsemax_kernel<<<NCOL, INPUT_DIM, 0, stream>>>(fsl, ft, lt, sel, scaleb);

    dim3 grid(BATCH / 16, NUM_TREES / 8);
    node_forward_kernel<<<grid, 128, 0, stream>>>(inputs, sel, scaleb, resp, out);
}